// SparseExecutionEngine_6717328851337
// MI455X (gfx1250) — compile-verified
//
#include <hip/hip_runtime.h>
#include <math.h>

typedef __attribute__((ext_vector_type(2))) float        v2f;
typedef __attribute__((ext_vector_type(8))) float        v8f;
typedef __attribute__((ext_vector_type(4))) unsigned int u32x4;
typedef __attribute__((ext_vector_type(4))) int          i32x4;
typedef __attribute__((ext_vector_type(8))) int          i32x8;

#define BB   4096
#define KK   64
#define DD   512
#define POOLN 1000000

// LDS layout (floats):
//   sel   [0,     32768)   64 rows x 512 f32 gathered tile (128 KB)
//   xl    [32768, 33280)   x[b] (2 KB)
//   prodp [33280, 33408)   2 x 64 partial dot products
//   wact  [33408, 33472)   64 weighted activations
#define SMEM_FLOATS 33472

__global__ __launch_bounds__(256) void sparse_engine_kernel(
    const float* __restrict__ x,
    const long long* __restrict__ indices,
    const float* __restrict__ weights,
    const float* __restrict__ pool,
    float* __restrict__ out)
{
  extern __shared__ float smem[];
  float* sel   = smem;
  float* xl    = smem + 32768;
  float* prodp = smem + 33280;
  float* wact  = smem + 33408;

  const int b    = blockIdx.x;
  const int tid  = threadIdx.x;
  const int wave = tid >> 5;
  const int lane = tid & 31;

  // ---- Stage x[b] into LDS with waves 1..7 (overlaps the TDM gather) ----
  if (wave != 0) {
    const float2* xsrc = (const float2*)(x + (size_t)b * DD);
    float2*       xdst = (float2*)xl;
    for (int i = tid - 32; i < DD / 2; i += 224)
      xdst[i] = xsrc[i];
  }

  // ---- Gather 64 scattered pool rows into LDS via TDM gather-mode DMA ----
#if __has_builtin(__builtin_amdgcn_tensor_load_to_lds)
  if (wave == 0) {
    const unsigned long long gaddr = (unsigned long long)pool;
    for (int grp = 0; grp < 8; ++grp) {
      // D# group 0: count=1, gather_mode=1, 32-bit indices, lds/global addr, type=2
      u32x4 g0;
      g0.x = 0xC0000001u;
      g0.y = (unsigned)(grp * 8 * DD * 4);                 // LDS byte offset: 8 rows/group
      g0.z = (unsigned)(gaddr & 0xFFFFFFFFull);
      g0.w = ((unsigned)(gaddr >> 32) & 0x01FFFFFFu) | 0x80000000u;
      // D# group 1: data_size=4B; tensor_dim0=512; tensor_dim1=POOL (OOB);
      //             tile_dim0=512; tile_dim1=8 valid indices; dim0_stride=512
      i32x8 g1;
      g1[0] = 0x00020000;
      g1[1] = (DD & 0xFFFF) << 16;                          // 0x02000000
      g1[2] = (int)(((POOLN & 0xFFFF) << 16) | (DD >> 16)); // 0x42400000
      g1[3] = (int)((DD << 16) | (POOLN >> 16));            // 0x0200000F
      g1[4] = 8;                                            // tile_dim1 = 8 valid indices
      g1[5] = DD;                                           // tensor_dim0_stride
      g1[6] = 0;
      g1[7] = 0;
      // D# groups 2/3: eight 32-bit row indices (uniform scalar loads)
      const long long* ip = indices + (size_t)b * KK + grp * 8;
      i32x4 g2, g3;
      g2[0] = (int)ip[0]; g2[1] = (int)ip[1]; g2[2] = (int)ip[2]; g2[3] = (int)ip[3];
      g3[0] = (int)ip[4]; g3[1] = (int)ip[5]; g3[2] = (int)ip[6]; g3[3] = (int)ip[7];
#if __clang_major__ >= 23
      i32x8 gz = {};
      __builtin_amdgcn_tensor_load_to_lds(g0, g1, g2, g3, gz, 0);
#else
      __builtin_amdgcn_tensor_load_to_lds(g0, g1, g2, g3, 0);
#endif
    }
    __builtin_amdgcn_s_wait_tensorcnt(0);
  }
#else
  // Fallback: cooperative vector-load gather (8 waves x 8 rows)
  for (int r = wave; r < KK; r += 8) {
    const long long row = indices[(size_t)b * KK + r];
    const float4* src = (const float4*)(pool + (size_t)row * DD);
    float4* dst = (float4*)(sel + r * DD);
    for (int i = lane; i < DD / 4; i += 32) dst[i] = src[i];
  }
#endif
  __syncthreads();

  // ---- Pass 1: prod[k] = dot(sel[k,:], x)  via V_WMMA_F32_16X16X4_F32 ----
  // Wave w handles k-tile (w&3) over D-half (w>>2). A = 16(k) x 4(d) tile of sel,
  // B = x[d..d+3] broadcast across all 16 N columns; accumulate over D in C.
  {
    const int ktile = wave & 3;
    const int dhalf = wave >> 2;
    const int k0    = ktile * 16;
    const int m     = lane & 15;
    const int c0    = (lane & 16) ? 2 : 0;   // inner-K pair select per A/B lane layout
    const float* ar = sel + (k0 + m) * DD + c0;
    const float* xr = xl + c0;
    const int dbase = dhalf * 256;

    v8f acc = {};
#pragma unroll 4
    for (int d = 0; d < 256; d += 4) {
      v2f a, bvec;
      a.x    = ar[dbase + d];
      a.y    = ar[dbase + d + 1];
      bvec.x = xr[dbase + d];
      bvec.y = xr[dbase + d + 1];
      acc = __builtin_amdgcn_wmma_f32_16x16x4_f32(
          false, a, false, bvec, (short)0, acc, false, false);
    }
    // Column N=0 lives in lane 0 (M=0..7) and lane 16 (M=8..15); every column equal.
    if ((lane & 15) == 0) {
      const int kb = k0 + ((lane >> 1) & 8);
#pragma unroll
      for (int r = 0; r < 8; ++r)
        prodp[dhalf * 64 + kb + r] = acc[r];
    }
  }
  __syncthreads();

  // ---- wact[k] = tanh(prod[k]) * weights[b,k] ----
  if (tid < KK) {
    const float p = prodp[tid] + prodp[64 + tid];
    wact[tid] = tanhf(p) * weights[(size_t)b * KK + tid];
  }
  __syncthreads();

  // ---- Pass 2: out[b,d] = x[d] + sum_k wact[k] * sel[k,d] ----
  // Each thread owns the contiguous pair d = 2*tid, 2*tid+1:
  // ds_load_b64 per k (conflict-free: dwords 2L,2L+1 span all 64 banks),
  // single b64 global store.
  {
    const int d2 = tid * 2;
    const float2* selrow = (const float2*)(sel + d2);   // stride DD/2 float2 per k
    float2 acc2; acc2.x = 0.0f; acc2.y = 0.0f;
#pragma unroll 8
    for (int k = 0; k < KK; ++k) {
      const float  wk = wact[k];
      const float2 s2 = selrow[k * (DD / 2)];
      acc2.x += wk * s2.x;
      acc2.y += wk * s2.y;
    }
    const float2 x2 = *(const float2*)(xl + d2);
    float2 o2; o2.x = x2.x + acc2.x; o2.y = x2.y + acc2.y;
    *(float2*)(out + (size_t)b * DD + d2) = o2;
  }
}

extern "C" void kernel_launch(void* const* d_in, const int* in_sizes, int n_in,
                              void* d_out, int out_size, void* d_ws, size_t ws_size,
                              hipStream_t stream) {
  const float*     x    = (const float*)d_in[0];
  const long long* idx  = (const long long*)d_in[1];   // int64 indices
  const float*     w    = (const float*)d_in[2];
  const float*     pool = (const float*)d_in[3];
  float*           out  = (float*)d_out;

  const size_t smem = SMEM_FLOATS * sizeof(float);     // ~131 KB < 320 KB/WGP
  (void)hipFuncSetAttribute((const void*)sparse_engine_kernel,
                            hipFuncAttributeMaxDynamicSharedMemorySize,
                            (int)smem);
  sparse_engine_kernel<<<dim3(BB), dim3(256), smem, stream>>>(x, idx, w, pool, out);
}